// AttentionHead_28321014350012
// MI455X (gfx1250) — compile-verified
//
#include <hip/hip_runtime.h>
#include <stdint.h>

// ---------------------------------------------------------------------------
// AttentionHead for MI455X (gfx1250, wave32, WMMA).
// Memory-bound (7.5 GFLOP vs ~110 MB HBM traffic -> ~4.5us floor @ 23.3TB/s):
// keep full fp32 precision via V_WMMA_F32_16X16X4_F32; K'/V' (1MB/batch)
// stay L2-resident; attention double-buffers them through LDS with the
// async-to-LDS engine (ASYNCcnt) so DMA overlaps WMMA+exp compute.
// ---------------------------------------------------------------------------

typedef float v2f __attribute__((ext_vector_type(2)));
typedef float v8f __attribute__((ext_vector_type(8)));
typedef int   v4i __attribute__((ext_vector_type(4)));

#define D_MODEL 1024
#define D_K     64
#define BATCH   4
#define SEQ     2048
#define ROWS    (BATCH * SEQ)   // 8192
#define NKB     (SEQ / 16)      // 128 key blocks

// Async global->LDS (ASYNCcnt-tracked). Probe-confirmed signature:
// (global v4i* src, local v4i* dst, imm offset, imm cpol).
#if defined(__AMDGCN__) && \
    __has_builtin(__builtin_amdgcn_global_load_async_to_lds_b128) && \
    __has_builtin(__builtin_amdgcn_s_wait_asynccnt)
#define USE_ASYNC_LDS 1
typedef __attribute__((address_space(1))) v4i GV4;
typedef __attribute__((address_space(3))) v4i LV4;
#else
#define USE_ASYNC_LDS 0
#endif

static __device__ __forceinline__ v8f wmma_f32_4(v2f a, v2f b, v8f c) {
    // D = A(16x4 f32) x B(4x16 f32) + C(16x16 f32)
    return __builtin_amdgcn_wmma_f32_16x16x4_f32(
        /*neg_a=*/false, a, /*neg_b=*/false, b,
        /*c_mod=*/(short)0, c, /*reuse_a=*/false, /*reuse_b=*/false);
}

// ===========================================================================
// Kernel 1: projection  Out[8192,64] = X[8192,1024] @ W[1024,64] + bias
// One wave: 16 rows x all 64 output cols (4 N-tiles). W staged transposed in
// LDS (stride 132 => 8B-aligned float2 B-fragments, conflict-free across both
// half-waves). X A-fragments are contiguous per-lane float2 global loads.
// grid = 64 blocks x 256 threads (8 waves): rowblock = blockIdx*8 + wave.
// ===========================================================================
#define K_CHUNK 128
#define WLDS_STRIDE (K_CHUNK + 4)   // 132: even (b64 align), 4c+k banks

__global__ __launch_bounds__(256)
void qkv_proj_kernel(const float* __restrict__ X,
                     const float* __restrict__ W,
                     const float* __restrict__ bias,
                     float* __restrict__ Out)
{
    __shared__ float wlds[D_K * WLDS_STRIDE];   // 64 x 132 floats = 33.8 KB

    const int lane = threadIdx.x & 31;
    const int wave = threadIdx.x >> 5;
    const int half = lane >> 4;    // 0: K+{0,1}, 1: K+{2,3}  (A/B VGPR halves)
    const int lm   = lane & 15;

    const int rowblock = blockIdx.x * 8 + wave;   // 0..511
    const int r0 = rowblock * 16;

    const v8f vzero = {0.f, 0.f, 0.f, 0.f, 0.f, 0.f, 0.f, 0.f};
    v8f acc[4];
#pragma unroll
    for (int nt = 0; nt < 4; ++nt) acc[nt] = vzero;

    // A-fragment source: row = r0 + lm (lanes 16..31 duplicate rows, K+2)
    const float* xrow = X + (size_t)(r0 + lm) * D_MODEL + 2 * half;

    for (int chunk = 0; chunk < D_MODEL / K_CHUNK; ++chunk) {
        __syncthreads();
        // Cooperative transposed stage of W[chunk*128 .. +128)[0..64) into LDS.
        const float* wsrc = W + chunk * (K_CHUNK * D_K);
#pragma unroll
        for (int i = 0; i < 8; ++i) {
            int f  = (threadIdx.x + i * 256) * 4;   // flat float index, 16B aligned
            int kk = f >> 6;                        // k within chunk
            int c  = f & 63;                        // output column
            float4 wv = *(const float4*)(wsrc + f);
            wlds[(c + 0) * WLDS_STRIDE + kk] = wv.x;
            wlds[(c + 1) * WLDS_STRIDE + kk] = wv.y;
            wlds[(c + 2) * WLDS_STRIDE + kk] = wv.z;
            wlds[(c + 3) * WLDS_STRIDE + kk] = wv.w;
        }
        __syncthreads();

        const float* xk = xrow + chunk * K_CHUNK;
#pragma unroll 4
        for (int ks = 0; ks < K_CHUNK / 4; ++ks) {
            // A 16x4: lane holds X[r0+lm][k0+2*half + {0,1}]
            v2f a = *(const v2f*)(xk + 4 * ks);
#pragma unroll
            for (int nt = 0; nt < 4; ++nt) {
                // B 4x16: lane holds W[k0+2*half + {0,1}][nt*16+lm]
                v2f b = *(const v2f*)(&wlds[(nt * 16 + lm) * WLDS_STRIDE + 4 * ks + 2 * half]);
                acc[nt] = wmma_f32_4(a, b, acc[nt]);
            }
        }
    }

    // C layout: VGPR r, lanes0-15 -> M=r, lanes16-31 -> M=r+8; N = lm.
#pragma unroll
    for (int nt = 0; nt < 4; ++nt) {
        float bv = bias[nt * 16 + lm];
#pragma unroll
        for (int r = 0; r < 8; ++r) {
            int row = r0 + r + 8 * half;
            Out[(size_t)row * D_K + nt * 16 + lm] = acc[nt][r] + bv;
        }
    }
}

// ===========================================================================
// Kernel 2: flash-style attention (unstabilized exp-softmax => no rescaling):
//   per 16-key block: K'/V' block staged into double-buffered LDS by the
//   async-to-LDS engine (next block's DMA overlaps current block's compute),
//   S = Q'K'^T/8, E = exp(S), accO += E @ V'block (4 WMMA tiles),
//   denom += E @ ones (1 WMMA tile), final out = accO / denom (identical
//   C-layout, pure register divide).
// E relayout C->A goes through a per-wave LDS buffer (stride 18: aligned
// float2 loads, conflict-free). K/V staging stride 68: 16B-aligned rows for
// B128 async writes, conflict-free K b64 fragment reads.
// grid = 64 blocks x 256 threads; 8 waves of a block share one batch and
// walk the key blocks in lockstep.
// ===========================================================================
#define KV_STRIDE 68   // 68*4 = 272 bytes: 16B-aligned rows; banks 4*row+col

__global__ __launch_bounds__(256)
void attention_kernel(const float* __restrict__ Qp,
                      const float* __restrict__ Kp,
                      const float* __restrict__ Vp,
                      float* __restrict__ Out)
{
    __shared__ float klds[2][16 * KV_STRIDE];   // K' block double buffer
    __shared__ float vlds[2][16 * KV_STRIDE];   // V' block double buffer
    __shared__ float elds[8][16 * 18];          // per-wave E tile, padded

    const int lane = threadIdx.x & 31;
    const int wave = threadIdx.x >> 5;
    const int half = lane >> 4;
    const int lm   = lane & 15;

    const int gw = blockIdx.x * 8 + wave;       // 0..511
    const int b  = gw >> 7;                     // batch (same for all 8 waves)
    const int qb = gw & 127;                    // query block in batch
    const size_t qrow0 = (size_t)b * SEQ + (size_t)qb * 16;
    const size_t krow0 = (size_t)b * SEQ;

    // Preload Q' A-fragments for the whole 16x64 query tile (16 k-steps).
    v2f aQ[16];
    const float* qptr = Qp + (qrow0 + lm) * D_K + 2 * half;
#pragma unroll
    for (int t = 0; t < 16; ++t) aQ[t] = *(const v2f*)(qptr + 4 * t);

    const v8f vzero = {0.f, 0.f, 0.f, 0.f, 0.f, 0.f, 0.f, 0.f};
    v8f accO[4];
#pragma unroll
    for (int nt = 0; nt < 4; ++nt) accO[nt] = vzero;
    v8f accD = vzero;
    const v2f bOne = {1.0f, 1.0f};

    // Cooperative staging map: thread -> one 16B chunk of the 16x64 block.
    const int srow = threadIdx.x >> 4;          // 0..15
    const int scol = (threadIdx.x & 15) * 4;    // 0,4,...,60
    const int sofs = srow * KV_STRIDE + scol;   // LDS element offset (16B aligned)
    const size_t gofs = (size_t)srow * D_K + scol;

    const float* Kg = Kp + krow0 * D_K;
    const float* Vg = Vp + krow0 * D_K;

    float* eb = &elds[wave][0];

#if USE_ASYNC_LDS
    // Pipeline prologue: start DMA of key block 0 into buffer 0.
    __builtin_amdgcn_global_load_async_to_lds_b128(
        (GV4*)(Kg + gofs), (LV4*)&klds[0][sofs], 0, 0);
    __builtin_amdgcn_global_load_async_to_lds_b128(
        (GV4*)(Vg + gofs), (LV4*)&vlds[0][sofs], 0, 0);
#endif

    for (int kb = 0; kb < NKB; ++kb) {
        const int cur = kb & 1;
        const int nxt = cur ^ 1;

        __syncthreads();   // all waves done reading the buffer we stage next
#if USE_ASYNC_LDS
        if (kb + 1 < NKB) {
            // Issue next block's DMA, then wait only for the current block
            // (in-order async completion: <=2 outstanding => current landed).
            const size_t ng = gofs + (size_t)(kb + 1) * 16 * D_K;
            __builtin_amdgcn_global_load_async_to_lds_b128(
                (GV4*)(Kg + ng), (LV4*)&klds[nxt][sofs], 0, 0);
            __builtin_amdgcn_global_load_async_to_lds_b128(
                (GV4*)(Vg + ng), (LV4*)&vlds[nxt][sofs], 0, 0);
            __builtin_amdgcn_s_wait_asynccnt(2);
        } else {
            __builtin_amdgcn_s_wait_asynccnt(0);
        }
#else
        {
            const size_t cg = gofs + (size_t)kb * 16 * D_K;
            float4 kv = *(const float4*)(Kg + cg);
            float4 vv = *(const float4*)(Vg + cg);
            *(float4*)&klds[cur][sofs] = kv;
            *(float4*)&vlds[cur][sofs] = vv;
        }
#endif
        __syncthreads();   // staged block visible to all 8 waves

        const float* kfrag = &klds[cur][lm * KV_STRIDE + 2 * half];
        const float* vfrag = &vlds[cur][2 * half * KV_STRIDE];

        // S = Q' @ K'^T  (K-dim 64 -> 16 WMMA steps); B(k,n)=K'[n][k] is a
        // contiguous float2 per lane from the staged row-major K' block.
        v8f s = vzero;
#pragma unroll
        for (int t = 0; t < 16; ++t) {
            v2f bK = *(const v2f*)(kfrag + 4 * t);
            s = wmma_f32_4(aQ[t], bK, s);
        }

        // E = exp(S / sqrt(64))  (faithful unstabilized softmax numerator)
        float e[8];
#pragma unroll
        for (int r = 0; r < 8; ++r) e[r] = __expf(s[r] * 0.125f);

        // C-layout -> A-layout relayout through per-wave LDS tile.
#pragma unroll
        for (int r = 0; r < 8; ++r) eb[(r + 8 * half) * 18 + lm] = e[r];
        __syncthreads();   // orders E stores before cross-lane E loads
        v2f aE[4];
#pragma unroll
        for (int t = 0; t < 4; ++t)
            aE[t] = *(const v2f*)(&eb[lm * 18 + 4 * t + 2 * half]);

        // accO += E @ V'block ; accD += E @ ones (row-sums, C-layout)
#pragma unroll
        for (int t = 0; t < 4; ++t) {
#pragma unroll
            for (int nt = 0; nt < 4; ++nt) {
                v2f bV;
                bV[0] = vfrag[(4 * t + 0) * KV_STRIDE + nt * 16 + lm];
                bV[1] = vfrag[(4 * t + 1) * KV_STRIDE + nt * 16 + lm];
                accO[nt] = wmma_f32_4(aE[t], bV, accO[nt]);
            }
            accD = wmma_f32_4(aE[t], bOne, accD);
        }
    }

    // Normalize: accD VGPR r holds denom[M] with the same (r,lane)->M map.
#pragma unroll
    for (int nt = 0; nt < 4; ++nt) {
#pragma unroll
        for (int r = 0; r < 8; ++r) {
            size_t row = qrow0 + r + 8 * half;
            Out[row * D_K + nt * 16 + lm] = accO[nt][r] / accD[r];
        }
    }
}

// ===========================================================================
// Launch: three projection launches (Q,K,V) into workspace, then attention.
// Workspace: 3 x 8192 x 64 floats = 6 MB.
// ===========================================================================
extern "C" void kernel_launch(void* const* d_in, const int* in_sizes, int n_in,
                              void* d_out, int out_size, void* d_ws, size_t ws_size,
                              hipStream_t stream) {
    const float* q  = (const float*)d_in[0];
    const float* k  = (const float*)d_in[1];
    const float* v  = (const float*)d_in[2];
    const float* Wq = (const float*)d_in[3];
    const float* bq = (const float*)d_in[4];
    const float* Wk = (const float*)d_in[5];
    const float* bk = (const float*)d_in[6];
    const float* Wv = (const float*)d_in[7];
    const float* bv = (const float*)d_in[8];

    float* ws = (float*)d_ws;
    float* Qp = ws;
    float* Kp = ws + (size_t)ROWS * D_K;
    float* Vp = ws + 2 * (size_t)ROWS * D_K;

    dim3 blk(256);
    qkv_proj_kernel<<<64, blk, 0, stream>>>(q, Wq, bq, Qp);
    qkv_proj_kernel<<<64, blk, 0, stream>>>(k, Wk, bk, Kp);
    qkv_proj_kernel<<<64, blk, 0, stream>>>(v, Wv, bv, Vp);
    attention_kernel<<<64, blk, 0, stream>>>(Qp, Kp, Vp, (float*)d_out);
}